// FirUpsample2D_13451837571633
// MI455X (gfx1250) — compile-verified
//
#include <hip/hip_runtime.h>

typedef __attribute__((ext_vector_type(2))) float v2f;
typedef __attribute__((ext_vector_type(4))) float v4f;
typedef __attribute__((ext_vector_type(8))) float v8f;
typedef __attribute__((ext_vector_type(4))) unsigned int v4u;
typedef __attribute__((ext_vector_type(4))) int v4i;
typedef __attribute__((ext_vector_type(8))) int v8i;

#define SP 144               // LDS pair-row stride: 128 cols + 16 pad (2*SP % 64 == 32 -> conflict-free halves)
#define SLAB_F (16 * SP * 2) // floats per slab buffer (16 k-pairs) = 4608
#define NSLAB 72             // 2304 / 32
#define GP_STRIDE 589824     // floats per parity in swizzled G = 2304*256

// 1-D parity mixing tables: MYTAB[parity][d (tap offset -1..1)][ky]
__device__ __constant__ float MYTAB[2][3][3] = {
    {{0.75f, 0.25f, 0.00f}, {0.25f, 0.75f, 0.75f}, {0.00f, 0.00f, 0.25f}},
    {{0.25f, 0.00f, 0.00f}, {0.75f, 0.75f, 0.25f}, {0.00f, 0.25f, 0.75f}}};

// ---------------------------------------------------------------------------
// Prep: G[p][k=t*256+ic][oc], stored K-pair swizzled so that the A slab in
// global memory is the exact LDS image the TDM copies:
//   flat = p*GP_STRIDE + ((k>>1)*256 + oc)*2 + (k&1)
// ---------------------------------------------------------------------------
__global__ __launch_bounds__(256) void prep_g(const float* __restrict__ w,
                                              float* __restrict__ G) {
  int idx = blockIdx.x * 256 + threadIdx.x;  // 4*9*256*256 = 2359296
  int oc = idx & 255;
  int ic = (idx >> 8) & 255;
  int rest = idx >> 16;  // p*9 + t
  int t = rest % 9;
  int p = rest / 9;
  if (p >= 4) return;
  int dy = t / 3, dx = t % 3;
  int py = p >> 1, px = p & 1;
  const float* wp = w + ((size_t)oc * 256 + ic) * 9;
  float acc = 0.0f;
#pragma unroll
  for (int ky = 0; ky < 3; ++ky)
#pragma unroll
    for (int kx = 0; kx < 3; ++kx)
      acc += MYTAB[py][dy][ky] * MYTAB[px][dx][kx] * wp[ky * 3 + kx];
  int kp = t * 128 + (ic >> 1);  // k-pair index of k = t*256+ic
  size_t o = (size_t)p * GP_STRIDE + (((size_t)kp * 256 + oc) << 1) + (ic & 1);
  G[o] = acc;
}

// ---------------------------------------------------------------------------
// Fused kernel: 4 parity GEMMs, M=256(oc) x N=32768(pix) x K=2304(ic*9 taps)
// WG tile 128x128, K-slab 32, double-buffered LDS. A slab moved by the
// Tensor Data Mover (TDM reproduces the padded LDS layout via pad fields).
// ---------------------------------------------------------------------------
__global__ __launch_bounds__(256) void fused_upconv(const float* __restrict__ h,
                                                    const float* __restrict__ G,
                                                    const float* __restrict__ bias,
                                                    float* __restrict__ out) {
  __shared__ float lsA[2][SLAB_F];
  __shared__ float lsB[2][SLAB_F];

  const int tid = threadIdx.x;
  const int lane = tid & 31;
  const int wave = tid >> 5;
  const int wm = wave & 3;   // oc sub-block (32)
  const int wn = wave >> 2;  // pixel sub-block (64)

  const int pixTile = blockIdx.x;     // 0..31
  const int ocTile = blockIdx.y & 1;  // 0..1
  const int par = blockIdx.y >> 1;    // 0..3
  const int n = blockIdx.z;           // 0..7
  const int py = par >> 1, px = par & 1;

  const int ocBase = ocTile * 128;
  const int pixBase = pixTile * 128;

  const float* Gp = G + (size_t)par * GP_STRIDE;
  const float* hn = h + (size_t)n * (256 * 4096);

  v8f acc[2][4];
#pragma unroll
  for (int mi = 0; mi < 2; ++mi)
#pragma unroll
    for (int ni = 0; ni < 4; ++ni)
#pragma unroll
      for (int j = 0; j < 8; ++j) acc[mi][ni][j] = 0.0f;

  const int lrow = lane & 15;
  float bR[16];  // staged B elements

  // ---- TDM: one descriptor copies the whole A slab global->LDS.
  // 2-D tile: 16 rows x 256 dwords, row stride 512 dwords; pad_enable with
  // pad_interval=7 (256 dwords) / pad_amount=31 (+32 dwords) recreates SP=144.
  auto tdm_issue_A = [&](int kbase, int buf) {
    if (wave == 0) {  // single issue per WG (TDM ignores EXEC)
      unsigned lds = (unsigned)(uintptr_t)&lsA[buf][0];
      unsigned long long ga =
          (unsigned long long)(uintptr_t)(Gp + (size_t)(kbase >> 1) * 512 + ocBase * 2);
      v4u g0;
      g0[0] = 1u;                                   // count=1, user descriptor
      g0[1] = lds;                                  // lds_addr (bytes)
      g0[2] = (unsigned)ga;                         // global_addr[31:0]
      g0[3] = (unsigned)((ga >> 32) & 0x01FFFFFFu)  // global_addr[56:32]
              | (2u << 30);                         // type=2 (image)
      v8i g1;
      g1[0] = 0x3FF20000;  // data_size=4B, pad_enable, pad_interval=7, pad_amount=31
      g1[1] = 0x01000000;  // tensor_dim0 = 256 (low16 in [31:16])
      g1[2] = 0x00100000;  // tensor_dim1 = 16  (low16 in [31:16])
      g1[3] = 0x01000000;  // tile_dim0 = 256   ([31:16])
      g1[4] = 16;          // tile_dim1 = 16
      g1[5] = 512;         // tensor_dim0_stride = 512 dwords
      g1[6] = 0;
      g1[7] = 0;
      v4i z4 = {0, 0, 0, 0};
      v8i z8 = {0, 0, 0, 0, 0, 0, 0, 0};
      __builtin_amdgcn_tensor_load_to_lds(g0, g1, z4, z4, z8, 0);
    }
  };
  auto tdm_wait = [&]() {
    if (wave == 0) __builtin_amdgcn_s_wait_tensorcnt(0);
  };

  // ---- B: global -> regs (predicated, zero halo) ----
  auto load_regsB = [&](int kbase) {
    const int t = kbase >> 8;  // 32 | 256 -> slab never crosses taps
    const int ic0 = kbase & 255;
    const int dy = t / 3 - 1;
    const int dx = t % 3 - 1;
#pragma unroll
    for (int i = 0; i < 16; ++i) {
      int f = tid + i * 256;
      int r = f >> 7;    // ic row in slab
      int pp = f & 127;  // pixel in tile
      int flat = pixBase + pp;
      int Y = flat >> 6, X = flat & 63;
      int Yp = Y + dy, Xp = X + dx;
      float v = 0.0f;
      if ((unsigned)Yp < 64u && (unsigned)Xp < 64u)
        v = hn[(size_t)(ic0 + r) * 4096 + Yp * 64 + Xp];
      bR[i] = v;
    }
  };
  auto store_ldsB = [&](int buf) {
#pragma unroll
    for (int i = 0; i < 16; ++i) {
      int f = tid + i * 256;
      int r = f >> 7;
      int pp = f & 127;
      lsB[buf][((r >> 1) * SP + pp) * 2 + (r & 1)] = bR[i];
    }
  };

  // ---- 8 WMMA K-steps per slab; each fragment = one aligned ds_load_b64 ----
  auto compute = [&](int buf) {
#pragma unroll
    for (int kp0 = 0; kp0 < 16; kp0 += 2) {
      const int kp = kp0 + (lane >> 4);  // lanes 0-15: K=k0,k0+1; 16-31: K=k0+2,k0+3
      v2f a[2], b[4];
#pragma unroll
      for (int mi = 0; mi < 2; ++mi) {
        int col = wm * 32 + mi * 16 + lrow;
        a[mi] = *(const v2f*)&lsA[buf][(kp * SP + col) * 2];
      }
#pragma unroll
      for (int ni = 0; ni < 4; ++ni) {
        int col = wn * 64 + ni * 16 + lrow;
        b[ni] = *(const v2f*)&lsB[buf][(kp * SP + col) * 2];
      }
#pragma unroll
      for (int mi = 0; mi < 2; ++mi)
#pragma unroll
        for (int ni = 0; ni < 4; ++ni)
          acc[mi][ni] = __builtin_amdgcn_wmma_f32_16x16x4_f32(
              false, a[mi], false, b[ni], (short)0, acc[mi][ni], false, false);
    }
  };

  // ---- pipeline: TDM(A,s+1) + regs(B,s+1) overlap compute(s) ----
  load_regsB(0);
  store_ldsB(0);
  tdm_issue_A(0, 0);
  tdm_wait();
  for (int s = 0; s < NSLAB; ++s) {
    __syncthreads();  // buf (s&1) fully written; buf ((s+1)&1) fully consumed
    if (s + 2 < NSLAB) {  // hint-prefetch A slab s+2 into L2 (global_prefetch_b8)
      const float* pf = Gp + (size_t)((s + 2) * 16 + (tid >> 4)) * 512 + ocBase * 2 +
                        (tid & 15) * 16;
      __builtin_prefetch(pf, 0, 0);
    }
    if (s + 1 < NSLAB) {
      tdm_issue_A((s + 1) * 32, (s + 1) & 1);
      load_regsB((s + 1) * 32);
    }
    compute(s & 1);
    if (s + 1 < NSLAB) {
      store_ldsB((s + 1) & 1);
      tdm_wait();  // wave 0: A slab landed before it signals next barrier
    }
  }

  // ---- epilogue: bias + stride-2 parity interleave into 128x128 output ----
  const int mhi = (lane >> 4) << 3;  // C layout: lanes 16-31 hold M = r+8
#pragma unroll
  for (int mi = 0; mi < 2; ++mi) {
#pragma unroll
    for (int r = 0; r < 8; ++r) {
      int oc = ocBase + wm * 32 + mi * 16 + mhi + r;
      float bv = bias[oc];
#pragma unroll
      for (int ni = 0; ni < 4; ++ni) {
        int flat = pixBase + wn * 64 + ni * 16 + lrow;
        int Y = flat >> 6, X = flat & 63;
        int y = 2 * Y + py, x = 2 * X + px;
        size_t o = (((size_t)(n * 256 + oc) * 128) + y) * 128 + x;
        out[o] = acc[mi][ni][r] + bv;
      }
    }
  }
}

extern "C" void kernel_launch(void* const* d_in, const int* in_sizes, int n_in,
                              void* d_out, int out_size, void* d_ws, size_t ws_size,
                              hipStream_t stream) {
  const float* h = (const float*)d_in[0];     // (8,256,64,64)
  const float* w = (const float*)d_in[1];     // (256,256,3,3)
  const float* bias = (const float*)d_in[2];  // (256,)
  float* out = (float*)d_out;                 // (8,256,128,128)
  float* G = (float*)d_ws;                    // 4*9*256*256 f32 = 9.44 MB scratch

  prep_g<<<9216, 256, 0, stream>>>(w, G);
  dim3 grid(32, 8, 8);  // pixTiles, ocTile*parity, batch
  fused_upconv<<<grid, 256, 0, stream>>>(h, G, bias, out);
}